// BankedLinear_36532991820308
// MI455X (gfx1250) — compile-verified
//
#include <hip/hip_runtime.h>

typedef __attribute__((ext_vector_type(2))) float v2f;
typedef __attribute__((ext_vector_type(8))) float v8f;

#define IN_F  128
#define OUT_F 128

// ---------------- binning: count / scan / fill ----------------

__global__ void k_zero(int* __restrict__ counts, int nb) {
    int i = blockIdx.x * blockDim.x + threadIdx.x;
    if (i < nb) counts[i] = 0;
}

__global__ void k_count(const int* __restrict__ sel, int NK, int* __restrict__ counts) {
    int i = blockIdx.x * blockDim.x + threadIdx.x;
    if (i < NK) atomicAdd(&counts[sel[i]], 1);
}

__global__ void k_scan(const int* __restrict__ counts, int* __restrict__ offsets,
                       int* __restrict__ cursor, int nb) {
    if (blockIdx.x == 0 && threadIdx.x == 0) {
        int s = 0;
        for (int b = 0; b < nb; ++b) { offsets[b] = s; cursor[b] = s; s += counts[b]; }
        offsets[nb] = s;
    }
}

__global__ void k_fill(const int* __restrict__ sel, int NK, int* __restrict__ cursor,
                       int* __restrict__ entries) {
    int i = blockIdx.x * blockDim.x + threadIdx.x;
    if (i < NK) {
        int p = atomicAdd(&cursor[sel[i]], 1);
        entries[p] = i;                      // encodes n*K + k
    }
}

// ---------------- bias init: out = sum_k bw * bias[sel_k] ----------------

__global__ void k_init_out(const float* __restrict__ bw, const int* __restrict__ sel,
                           const float* __restrict__ bias, float* __restrict__ out,
                           int N, int K) {
    int idx = blockIdx.x * blockDim.x + threadIdx.x;
    if (idx >= N * OUT_F) return;
    int n = idx >> 7, c = idx & (OUT_F - 1);
    float acc = 0.0f;
    for (int k = 0; k < K; ++k)
        acc += bw[n * K + k] * bias[sel[n * K + k] * OUT_F + c];
    out[idx] = acc;
}

// ---------------- grouped GEMM per bank with fp32 WMMA ----------------
// block = 128 threads (4 waves). Wave w owns output columns [w*32, w*32+32).
// Each (bank, tile) processes 16 gathered token rows: Y[16x128] = Xg[16x128] @ W[b].

__global__ __launch_bounds__(128) void k_bank_gemm(
    const float* __restrict__ X, const float* __restrict__ Wt,
    const float* __restrict__ BW, const int* __restrict__ offsets,
    const int* __restrict__ entries, float* __restrict__ out, int K)
{
    __shared__ float xs[16][132];   // 132-float row stride: conflict-free A-frag reads
    __shared__ int   tok[16];
    __shared__ float bwv[16];

    const int bank   = blockIdx.x;
    const int beg    = offsets[bank];
    const int count  = offsets[bank + 1] - beg;
    const int ntiles = (count + 15) >> 4;
    const float* Wb  = Wt + (size_t)bank * IN_F * OUT_F;

    const int tid  = threadIdx.x;
    const int lane = tid & 31;
    const int wave = tid >> 5;
    const int hi   = lane >> 4;     // lane half selects K pair / M half
    const int lo   = lane & 15;
    const int ncol = wave * 32;

    for (int t = blockIdx.y; t < ntiles; t += gridDim.y) {
        const int base  = beg + t * 16;
        const int rem   = count - t * 16;
        const int valid = (rem < 16) ? rem : 16;

        if (tid < 16) {
            int e = entries[base + ((tid < valid) ? tid : 0)];  // dummy = row 0
            tok[tid] = e / K;
            bwv[tid] = (tid < valid) ? BW[e] : 0.0f;
        }
        __syncthreads();

        // gather 16 token rows of X into LDS: 8 threads/row, 16 floats/thread
        {
            const int r  = tid >> 3;
            const int cb = (tid & 7) * 16;
            const float4* src = (const float4*)(X + (size_t)tok[r] * IN_F + cb);
            float4* dst = (float4*)&xs[r][cb];
            dst[0] = src[0]; dst[1] = src[1]; dst[2] = src[2]; dst[3] = src[3];
        }
        __syncthreads();

        v8f acc0 = {}; v8f acc1 = {};
        for (int k0 = 0; k0 < IN_F; k0 += 4) {
            const int kb = k0 + 2 * hi;
            // A 16x4 f32 frag: lane(lo,hi) holds X[m=lo][kb], X[m=lo][kb+1]
            v2f a = *(const v2f*)&xs[lo][kb];
            // B 4x16 f32 frags: lane(lo,hi) holds W[kb][n], W[kb+1][n]
            const float* wp = Wb + (size_t)kb * OUT_F + ncol + lo;
            v2f b0, b1;
            b0.x = wp[0];   b0.y = wp[OUT_F];
            b1.x = wp[16];  b1.y = wp[OUT_F + 16];
            acc0 = __builtin_amdgcn_wmma_f32_16x16x4_f32(
                       false, a, false, b0, (short)0, acc0, false, false);
            acc1 = __builtin_amdgcn_wmma_f32_16x16x4_f32(
                       false, a, false, b1, (short)0, acc1, false, false);
        }

        // epilogue: out[tok[m]] += bw[m] * Y  (exactly K=2 commutative adds/elem)
        #pragma unroll
        for (int r = 0; r < 8; ++r) {
            const int m = r + 8 * hi;           // C/D layout: VGPR r, lane half -> M
            if (m < valid) {
                const float w = bwv[m];
                float* orow = out + (size_t)tok[m] * OUT_F + ncol + lo;
                atomicAdd(&orow[0],  w * acc0[r]);
                atomicAdd(&orow[16], w * acc1[r]);
            }
        }
        __syncthreads();
    }
}

// ---------------- launcher ----------------

extern "C" void kernel_launch(void* const* d_in, const int* in_sizes, int n_in,
                              void* d_out, int out_size, void* d_ws, size_t ws_size,
                              hipStream_t stream) {
    const float* X    = (const float*)d_in[0];   // [N, 128]
    const float* BW   = (const float*)d_in[1];   // [N, K]
    const int*   SEL  = (const int*)  d_in[2];   // [N, K]
    const float* Wt   = (const float*)d_in[3];   // [NB, 128, 128]
    const float* bias = (const float*)d_in[4];   // [NB, 128]
    float* out = (float*)d_out;

    const int N  = in_sizes[0] / IN_F;
    const int K  = in_sizes[1] / N;
    const int NK = N * K;
    const int NB = in_sizes[4] / OUT_F;

    // workspace layout (ints): [0,64) counts, [64,129) offsets, [192,256) cursor,
    // [256, 256+NK) entries
    int* wsI     = (int*)d_ws;
    int* counts  = wsI;
    int* offsets = wsI + 64;
    int* cursor  = wsI + 192;
    int* entries = wsI + 256;

    k_zero <<<1, 64, 0, stream>>>(counts, NB);
    k_count<<<(NK + 255) / 256, 256, 0, stream>>>(SEL, NK, counts);
    k_scan <<<1, 1, 0, stream>>>(counts, offsets, cursor, NB);
    k_fill <<<(NK + 255) / 256, 256, 0, stream>>>(SEL, NK, cursor, entries);
    k_init_out<<<(N * OUT_F + 255) / 256, 256, 0, stream>>>(BW, SEL, bias, out, N, K);

    dim3 grid(NB, 32);              // y-slots strided over per-bank tiles
    k_bank_gemm<<<grid, 128, 0, stream>>>(X, Wt, BW, offsets, entries, out, K);
}